// PartitionedNormalization_87995289960768
// MI455X (gfx1250) — compile-verified
//
#include <hip/hip_runtime.h>
#include <hip/hip_bf16.h>

typedef __attribute__((ext_vector_type(2))) float v2f;
typedef __attribute__((ext_vector_type(8))) float v8f;

#define EPS_BN 1e-3f
#define D_DIM 512
#define ND_DIM 4
#define STAT_ROWS 256   // rows reduced per block in the stats kernel

// ---- workspace layout (bytes) ----
#define OFF_DOM 0          // int dom[B]            (65536*4 = 262144)
#define OFF_CNT 262144     // uint counts[4]
#define OFF_SUM 262400     // float sums[4*512]
#define OFF_SQ  270592     // float sqsums[4*512]
#define OFF_INV 278784     // float inv[4*512]
#define OFF_SHF 286976     // float shift[4*512]

// Kernel 0: zero the accumulator region (sums, sqsums, counts).
__global__ __launch_bounds__(256) void pn_zero_kernel(float* sums, float* sqs,
                                                      unsigned* counts) {
  int i = blockIdx.x * 256 + threadIdx.x;
  if (i < 2048) { sums[i] = 0.0f; sqs[i] = 0.0f; }
  if (i < ND_DIM) counts[i] = 0u;
}

// Kernel 1: per-row domain id (argmax of one-hot) + histogram of counts.
__global__ __launch_bounds__(256) void pn_dom_kernel(const float* __restrict__ ind,
                                                     int* __restrict__ dom,
                                                     unsigned* __restrict__ counts) {
  __shared__ unsigned hist[ND_DIM];
  if (threadIdx.x < ND_DIM) hist[threadIdx.x] = 0u;
  __syncthreads();
  int r = blockIdx.x * 256 + threadIdx.x;
  const float4 v = *(const float4*)(ind + (size_t)r * ND_DIM);
  int d = 0;
  float best = v.x;
  if (v.y > best) { best = v.y; d = 1; }
  if (v.z > best) { best = v.z; d = 2; }
  if (v.w > best) { d = 3; }
  dom[r] = d;
  atomicAdd(&hist[d], 1u);
  __syncthreads();
  if (threadIdx.x < ND_DIM) atomicAdd(&counts[threadIdx.x], hist[threadIdx.x]);
}

// Kernel 2: per-domain sum/sqsum via V_WMMA_F32_16X16X4_F32.
//   D(m,n) += sum_k  A(m,k) * B(k,n)
//   A(m,k) = X[rowbase+k][cb+m]   (f32 A layout: lanes 0-15 = M, lane-half x VGPR = K)
//   B(k,n) = (dom[rowbase+k] == n)  for n < 4, else 0
// => D(m,n) = sum over rows of X[r][cb+m] * (dom_r == n)  == sums[n][cb+m] partial.
// Accumulator stays in 8 VGPRs across STAT_ROWS rows; block-end atomics merge.
__global__ __launch_bounds__(256) void pn_stats_kernel(const float* __restrict__ x,
                                                       const int* __restrict__ dom,
                                                       float* __restrict__ sums,
                                                       float* __restrict__ sqs) {
  __shared__ int sdom[STAT_ROWS];
  const int tid = threadIdx.x;
  sdom[tid] = dom[blockIdx.y * STAT_ROWS + tid];
  __syncthreads();

  const int lane   = tid & 31;
  const int wave   = tid >> 5;
  const int laneHi = lane >> 4;      // 0: lanes 0-15 (K=0,1), 1: lanes 16-31 (K=2,3)
  const int m      = lane & 15;      // A: column-within-group; B/D: N index
  const int cb     = blockIdx.x * 128 + wave * 16;

  const float* p = x + ((size_t)blockIdx.y * STAT_ROWS + 2 * laneHi) * D_DIM + cb + m;

  v8f dsum = {};
  v8f dsq  = {};

  for (int r0 = 0; r0 < STAT_ROWS; r0 += 4) {
    v2f a;
    a.x = p[(size_t)r0 * D_DIM];            // row r0 + 2*laneHi
    a.y = p[(size_t)r0 * D_DIM + D_DIM];    // row r0 + 2*laneHi + 1
    const int k0 = r0 + 2 * laneHi;
    v2f b;
    b.x = (m == sdom[k0])     ? 1.0f : 0.0f;
    b.y = (m == sdom[k0 + 1]) ? 1.0f : 0.0f;
    v2f a2 = a * a;
    dsum = __builtin_amdgcn_wmma_f32_16x16x4_f32(false, a,  false, b, (short)0,
                                                 dsum, false, false);
    dsq  = __builtin_amdgcn_wmma_f32_16x16x4_f32(false, a2, false, b, (short)0,
                                                 dsq,  false, false);
  }

  // D layout: VGPR j, lanes 0-15 -> (M=j, N=lane); lanes 16-31 -> (M=8+j, N=lane-16)
  const int n = lane & 15;
  if (n < ND_DIM) {
#pragma unroll
    for (int j = 0; j < 8; ++j) {
      const int col = cb + j + 8 * laneHi;
      atomicAdd(&sums[n * D_DIM + col], dsum[j]);
      atomicAdd(&sqs [n * D_DIM + col], dsq[j]);
    }
  }
}

// Kernel 3: finalize per-domain affine params (4 x 512, tiny).
__global__ __launch_bounds__(256) void pn_finalize_kernel(const float* __restrict__ sums,
                                                          const float* __restrict__ sqs,
                                                          const unsigned* __restrict__ counts,
                                                          const float* __restrict__ gamma,
                                                          const float* __restrict__ beta,
                                                          float* __restrict__ inv,
                                                          float* __restrict__ shift) {
  int i = blockIdx.x * 256 + threadIdx.x;
  if (i >= ND_DIM * D_DIM) return;
  int nd = i >> 9;
  unsigned c = counts[nd];
  float cnt = (c > 0u) ? (float)c : 1.0f;
  float rc = 1.0f / cnt;
  float mean = sums[i] * rc;
  float var  = sqs[i] * rc - mean * mean;
  float iv   = gamma[i] * rsqrtf(var + EPS_BN);
  inv[i]   = iv;
  shift[i] = beta[i] - mean * iv;
}

// Kernel 4: y = x*inv[dom] + shift[dom], float4 (b128) streaming pass.
__global__ __launch_bounds__(256) void pn_norm_kernel(const float* __restrict__ x,
                                                      const int* __restrict__ dom,
                                                      const float* __restrict__ inv,
                                                      const float* __restrict__ shift,
                                                      float* __restrict__ out) {
  size_t e = ((size_t)blockIdx.x * 256 + threadIdx.x) * 4;
  int row = (int)(e >> 9);
  int col = (int)(e & (D_DIM - 1));
  int d = dom[row];
  const float4 xv = *(const float4*)(x + e);
  const float4 iv = *(const float4*)(inv + d * D_DIM + col);
  const float4 sv = *(const float4*)(shift + d * D_DIM + col);
  float4 o;
  o.x = fmaf(xv.x, iv.x, sv.x);
  o.y = fmaf(xv.y, iv.y, sv.y);
  o.z = fmaf(xv.z, iv.z, sv.z);
  o.w = fmaf(xv.w, iv.w, sv.w);
  *(float4*)(out + e) = o;
}

extern "C" void kernel_launch(void* const* d_in, const int* in_sizes, int n_in,
                              void* d_out, int out_size, void* d_ws, size_t ws_size,
                              hipStream_t stream) {
  const float* x     = (const float*)d_in[0];   // [B, 512] fp32
  const float* ind   = (const float*)d_in[1];   // [B, 4]   fp32 one-hot
  const float* gamma = (const float*)d_in[2];   // [4, 512]
  const float* beta  = (const float*)d_in[3];   // [4, 512]
  float* out = (float*)d_out;

  const int B = in_sizes[0] / D_DIM;            // 65536

  char* ws = (char*)d_ws;
  int*      dom    = (int*)(ws + OFF_DOM);
  unsigned* counts = (unsigned*)(ws + OFF_CNT);
  float*    sums   = (float*)(ws + OFF_SUM);
  float*    sqs    = (float*)(ws + OFF_SQ);
  float*    inv    = (float*)(ws + OFF_INV);
  float*    shift  = (float*)(ws + OFF_SHF);

  // 0) zero accumulators (ws is poisoned, not re-zeroed between replays)
  pn_zero_kernel<<<8, 256, 0, stream>>>(sums, sqs, counts);

  // 1) domain ids + counts
  pn_dom_kernel<<<B / 256, 256, 0, stream>>>(ind, dom, counts);

  // 2) WMMA segment reduction: sums/sqsums
  dim3 sgrid(D_DIM / 128, B / STAT_ROWS);       // (4, 256)
  pn_stats_kernel<<<sgrid, 256, 0, stream>>>(x, dom, sums, sqs);

  // 3) finalize affine params
  pn_finalize_kernel<<<(ND_DIM * D_DIM + 255) / 256, 256, 0, stream>>>(
      sums, sqs, counts, gamma, beta, inv, shift);

  // 4) streaming normalize (x likely still L2-resident: 134 MB < 192 MB L2)
  pn_norm_kernel<<<(B * (D_DIM / 4)) / 256, 256, 0, stream>>>(x, dom, inv, shift, out);
}